// RayTracing_26456998544124
// MI455X (gfx1250) — compile-verified
//
#include <hip/hip_runtime.h>
#include <math.h>

// Ray tracing (IDR-style sphere tracing + dense ray sampler + secant) for
// MI455X / gfx1250. Pure per-ray fp32 VALU/TRANS workload: one fused kernel,
// one thread per ray, wave32 divergence culling of the expensive sampler.

#define R2           1.0f       // bounding sphere radius^2 (R=1)
#define SDF_THR      5e-05f
#define LS_STEP      0.5f       // (1 - LINE_SEARCH_STEP) with 1 line-step iter
#define TRACE_ITERS  10
#define NSTEPS       100
#define NSECANT      8

__device__ __forceinline__ float sdf_eval(float px, float py, float pz,
                                          float scx, float scy, float scz,
                                          float rad) {
    float ax = px - scx, ay = py - scy, az = pz - scz;
    return sqrtf(ax * ax + ay * ay + az * az + 1e-12f) - rad;
}

__global__ void __launch_bounds__(256)
raytrace_fused(const float* __restrict__ cam_loc,     // [B,3]
               const float* __restrict__ dirs,        // [N,3]
               const float* __restrict__ sdf_center,  // [3]
               const float* __restrict__ sdf_radius,  // [1]
               float* __restrict__ out,               // [3N | N | N]
               int N, int P) {
    int i = blockIdx.x * blockDim.x + threadIdx.x;
    if (i >= N) return;

    // gfx1250 prefetch of the ray-direction stream (global_prefetch_b8)
    __builtin_prefetch(&dirs[(size_t)i * 3], 0, 0);

    int b = (int)((unsigned)i / (unsigned)P);
    float cx = cam_loc[b * 3 + 0];
    float cy = cam_loc[b * 3 + 1];
    float cz = cam_loc[b * 3 + 2];
    float dx = dirs[(size_t)i * 3 + 0];
    float dy = dirs[(size_t)i * 3 + 1];
    float dz_ = dirs[(size_t)i * 3 + 2];
    float scx = sdf_center[0], scy = sdf_center[1], scz = sdf_center[2];
    float rad = sdf_radius[0];

    // ---- bounding-sphere intersection ----
    float rcd = dx * cx + dy * cy + dz_ * cz;
    float c2  = cx * cx + cy * cy + cz * cz;
    float under = rcd * rcd - (c2 - R2);
    bool mask_i = under > 0.0f;
    float sq = sqrtf(mask_i ? under : 0.0f);
    float acc_s = mask_i ? fmaxf(-sq - rcd, 0.0f) : 0.0f;
    float acc_e = mask_i ? fmaxf( sq - rcd, 0.0f) : 0.0f;

    // ---- sphere tracing from both ends ----
    bool unfin_s = mask_i, unfin_e = mask_i;
    float next_s = mask_i ? sdf_eval(cx + acc_s * dx, cy + acc_s * dy, cz + acc_s * dz_,
                                     scx, scy, scz, rad) : 0.0f;
    float next_e = mask_i ? sdf_eval(cx + acc_e * dx, cy + acc_e * dy, cz + acc_e * dz_,
                                     scx, scy, scz, rad) : 0.0f;

    for (int it = 0; it < TRACE_ITERS; ++it) {
        float curr_s = unfin_s ? next_s : 0.0f;
        curr_s = (curr_s <= SDF_THR) ? 0.0f : curr_s;
        float curr_e = unfin_e ? next_e : 0.0f;
        curr_e = (curr_e <= SDF_THR) ? 0.0f : curr_e;
        unfin_s = unfin_s && (curr_s > SDF_THR);
        unfin_e = unfin_e && (curr_e > SDF_THR);
        acc_s += curr_s;
        acc_e -= curr_e;
        next_s = unfin_s ? sdf_eval(cx + acc_s * dx, cy + acc_s * dy, cz + acc_s * dz_,
                                    scx, scy, scz, rad) : 0.0f;
        next_e = unfin_e ? sdf_eval(cx + acc_e * dx, cy + acc_e * dy, cz + acc_e * dz_,
                                    scx, scy, scz, rad) : 0.0f;
        // line-search back-off (1 iter, step 0.5)
        if (next_s < 0.0f) {
            acc_s -= LS_STEP * curr_s;
            next_s = sdf_eval(cx + acc_s * dx, cy + acc_s * dy, cz + acc_s * dz_,
                              scx, scy, scz, rad);
        }
        if (next_e < 0.0f) {
            acc_e += LS_STEP * curr_e;
            next_e = sdf_eval(cx + acc_e * dx, cy + acc_e * dy, cz + acc_e * dz_,
                              scx, scy, scz, rad);
        }
        bool ok = acc_s < acc_e;
        unfin_s = unfin_s && ok;
        unfin_e = unfin_e && ok;
    }
    // final mask refresh (break iteration of the torch while-loop)
    bool sampler_mask = unfin_s && (next_s > SDF_THR);
    bool net_obj_mask = acc_s < acc_e;

    // defaults (non-sampler path): pts_s = cam + acc_s*dir (loop writes it
    // unmasked each iteration, so this holds for all rays incl. non-intersect)
    float opx = cx + acc_s * dx;
    float opy = cy + acc_s * dy;
    float opz = cz + acc_s * dz_;
    float odist = acc_s;
    float omask = net_obj_mask ? 1.0f : 0.0f;

    if (sampler_mask) {
        // ---- dense ray sampler: 100 steps, first sign-flip via argmin trick ----
        const float inv99 = 1.0f / 99.0f;
        float smin = acc_s;
        float dzr  = acc_e - acc_s;     // (smax - smin)
        float best = 3.4e38f;
        int ind = 0;
        float sdf_hi = 0.0f;
        #pragma unroll 5
        for (int s = 0; s < NSTEPS; ++s) {
            float t = (float)s * inv99;
            float z = smin + t * dzr;
            float v = sdf_eval(cx + z * dx, cy + z * dy, cz + z * dz_,
                               scx, scy, scz, rad);
            float sg  = (v > 0.0f) ? 1.0f : ((v < 0.0f) ? -1.0f : 0.0f);
            float tmp = sg * (float)(NSTEPS - s);   // arange(100,0,-1)
            if (tmp < best) { best = tmp; ind = s; sdf_hi = v; }  // first min
        }
        int lo = (ind == 0) ? (NSTEPS - 1) : (ind - 1);  // torch-style -1 wrap
        float z_hi = smin + (float)ind * inv99 * dzr;
        float z_lo = smin + (float)lo  * inv99 * dzr;
        float sdf_lo = sdf_eval(cx + z_lo * dx, cy + z_lo * dy, cz + z_lo * dz_,
                                scx, scy, scz, rad);
        bool net_surface = sdf_hi < 0.0f;

        if (net_surface) {
            // ---- secant root finding ----
            float zl = z_lo, zh = z_hi, sl = sdf_lo, sh = sdf_hi;
            float den = sh - sl; den = (den == 0.0f) ? 1.0f : den;
            float zp = -sl * ((zh - zl) / den) + zl;
            #pragma unroll
            for (int k = 0; k < NSECANT; ++k) {
                float sm = sdf_eval(cx + zp * dx, cy + zp * dy, cz + zp * dz_,
                                    scx, scy, scz, rad);
                if (sm > 0.0f)      { zl = zp; sl = sm; }
                else if (sm < 0.0f) { zh = zp; sh = sm; }
                den = sh - sl; den = (den == 0.0f) ? 1.0f : den;
                zp = -sl * ((zh - zl) / den) + zl;
            }
            opx = cx + zp * dx; opy = cy + zp * dy; opz = cz + zp * dz_;
            odist = zp;
            omask = 1.0f;           // sampler_mask & net_surface
        } else {
            // fallback: p_ind = pts[row, ind], z_high (pre-secant)
            opx = cx + z_hi * dx; opy = cy + z_hi * dy; opz = cz + z_hi * dz_;
            odist = z_hi;
            omask = 0.0f;           // sampler_mask & net_surface == false
        }
    }

    // outputs concatenated flat: points [N,3] | mask [N] | dists [N]
    out[(size_t)i * 3 + 0] = opx;
    out[(size_t)i * 3 + 1] = opy;
    out[(size_t)i * 3 + 2] = opz;
    out[(size_t)3 * N + i] = omask;
    out[(size_t)4 * N + i] = odist;
}

extern "C" void kernel_launch(void* const* d_in, const int* in_sizes, int n_in,
                              void* d_out, int out_size, void* d_ws, size_t ws_size,
                              hipStream_t stream) {
    const float* cam_loc    = (const float*)d_in[0];  // [B,3]
    const float* dirs       = (const float*)d_in[1];  // [B,P,3]
    // d_in[2] object_mask: unused in eval-mode forward
    const float* sdf_center = (const float*)d_in[3];  // [3]
    const float* sdf_radius = (const float*)d_in[4];  // [1]
    float* out = (float*)d_out;

    int B = in_sizes[0] / 3;
    int N = in_sizes[2];          // B*P rays
    int P = N / (B > 0 ? B : 1);

    dim3 block(256);              // 8 wave32 waves per block
    dim3 grid((N + 255) / 256);
    raytrace_fused<<<grid, block, 0, stream>>>(cam_loc, dirs, sdf_center,
                                               sdf_radius, out, N, P);
}